// VanillaGCNStack_62036507623919
// MI455X (gfx1250) — compile-verified
//
#include <hip/hip_runtime.h>
#include <hip/hip_bf16.h>

#define N_NODES  100000
#define N_EDGES  3200000
#define N_GRAPHS 256
#define HID      32
#define IN_DIM   6

typedef __attribute__((ext_vector_type(2))) float v2f;
typedef __attribute__((ext_vector_type(8))) float v8f;

// ---------------------------------------------------------------- utilities
__global__ void fill_zero_kernel(float* __restrict__ p, int n) {
  int i = blockIdx.x * blockDim.x + threadIdx.x;
  if (i < n) p[i] = 0.0f;
}

// --------------------------------------------------- edge weights + degree
__global__ void edge_weight_kernel(const float* __restrict__ pos,
                                   const int* __restrict__ row,
                                   const int* __restrict__ col,
                                   const float* __restrict__ s1p,
                                   const float* __restrict__ s2p,
                                   float* __restrict__ ew,
                                   float* __restrict__ deg) {
  int e = blockIdx.x * blockDim.x + threadIdx.x;
  if (e >= N_EDGES) return;
  int r = row[e], c = col[e];
  float s1 = s1p[0], s2 = s2p[0];
  const float* pr = pos + (size_t)r * 6;
  const float* pc = pos + (size_t)c * 6;
  float d0 = pr[0] - pc[0], d1 = pr[1] - pc[1], d2 = pr[2] - pc[2];
  float D = d0 * d0 + d1 * d1 + d2 * d2;
  float t = 1.0f - (pr[3] * pc[3] + pr[4] * pc[4] + pr[5] * pc[5]);
  float w = __expf(-(D * s1 * s1 + t * t * s2 * s2));
  ew[e] = w;
  atomicAdd(&deg[c], w);
}

__global__ void dinv_kernel(const float* __restrict__ deg, float* __restrict__ dinv) {
  int i = blockIdx.x * blockDim.x + threadIdx.x;
  if (i >= N_NODES) return;
  dinv[i] = rsqrtf(deg[i] + 1.0f);   // deg+1 >= 1, so the "where" guard is moot
}

// fold dinv[row]*ew*dinv[col] into one per-edge coefficient (in-place on ew)
__global__ void coef_kernel(const int* __restrict__ row,
                            const int* __restrict__ col,
                            const float* __restrict__ dinv,
                            float* __restrict__ ew) {
  int e = blockIdx.x * blockDim.x + threadIdx.x;
  if (e >= N_EDGES) return;
  ew[e] = dinv[row[e]] * ew[e] * dinv[col[e]];
}

// -------------------------------------------------- layer-1 GEMM (K=6, VALU)
__global__ void gemm_in_kernel(const float* __restrict__ x,
                               const float* __restrict__ W,
                               float* __restrict__ xw) {
  int gid = blockIdx.x * blockDim.x + threadIdx.x;
  if (gid >= N_NODES * HID) return;
  int node = gid >> 5;
  int f    = gid & 31;
  float acc = 0.0f;
#pragma unroll
  for (int k = 0; k < IN_DIM; ++k)
    acc += x[node * IN_DIM + k] * W[k * HID + f];
  xw[gid] = acc;
}

// ------------------------------------- 32x32 GEMM via V_WMMA_F32_16X16X4_F32
// Each wave32 owns one 16-row tile; K=32 consumed in 8 WMMA steps for both
// 16-column halves of the 32-wide output.  100000 rows = 6250 exact tiles.
__global__ __launch_bounds__(128) void gemm_hid_wmma(const float* __restrict__ H,
                                                     const float* __restrict__ W,
                                                     float* __restrict__ XW) {
  int wave = blockIdx.x * (blockDim.x >> 5) + (threadIdx.x >> 5);
  int lane = threadIdx.x & 31;
  int row0 = wave * 16;
  if (row0 >= N_NODES) return;          // uniform per wave -> EXEC stays all-1s
  int grp = lane >> 4;                  // 0: K lanes 0/1 ; 1: K lanes 2/3
  int m   = lane & 15;
  v8f c0 = {0.f,0.f,0.f,0.f,0.f,0.f,0.f,0.f};
  v8f c1 = {0.f,0.f,0.f,0.f,0.f,0.f,0.f,0.f};
  const float* hrow = H + (size_t)(row0 + m) * HID;
#pragma unroll
  for (int k0 = 0; k0 < HID; k0 += 4) {
    int kk = k0 + 2 * grp;
    v2f a;  a.x  = hrow[kk];                 a.y  = hrow[kk + 1];
    v2f b0; b0.x = W[kk * HID + m];          b0.y = W[(kk + 1) * HID + m];
    v2f b1; b1.x = W[kk * HID + 16 + m];     b1.y = W[(kk + 1) * HID + 16 + m];
    c0 = __builtin_amdgcn_wmma_f32_16x16x4_f32(false, a, false, b0, (short)0, c0, false, false);
    c1 = __builtin_amdgcn_wmma_f32_16x16x4_f32(false, a, false, b1, (short)0, c1, false, false);
  }
#pragma unroll
  for (int r = 0; r < 8; ++r) {
    int M = r + 8 * grp;                // C/D layout: VGPR r, lane-half adds 8
    XW[(size_t)(row0 + M) * HID + m]      = c0[r];
    XW[(size_t)(row0 + M) * HID + 16 + m] = c1[r];
  }
}

// --------------------------------------------- edge scatter (wave per edge)
__global__ void scatter_kernel(const int* __restrict__ row,
                               const int* __restrict__ col,
                               const float* __restrict__ coef,
                               const float* __restrict__ xw,
                               float* __restrict__ out) {
  int gid = blockIdx.x * blockDim.x + threadIdx.x;   // < N_EDGES*32 = 102.4M
  int e = gid >> 5;                                  // same edge for all 32 lanes
  int f = gid & 31;
  if (e >= N_EDGES) return;
  float v = coef[e] * xw[(size_t)row[e] * HID + f];  // coalesced 128B gather
  atomicAdd(&out[(size_t)col[e] * HID + f], v);      // coalesced 128B atomic burst
}

// ------------------------------- self-loop + bias (+ BN + ReLU for L1..L3)
__global__ void finalize_kernel(const float* __restrict__ conv,
                                const float* __restrict__ xw,
                                const float* __restrict__ dinv,
                                const float* __restrict__ b,
                                const float* __restrict__ g,
                                const float* __restrict__ be,
                                int do_bn,
                                float* __restrict__ out) {
  int gid = blockIdx.x * blockDim.x + threadIdx.x;
  if (gid >= N_NODES * HID) return;
  int node = gid >> 5;
  int f    = gid & 31;
  float di = dinv[node];
  float v  = conv[gid] + di * di * xw[gid] + b[f];
  if (do_bn) {
    v = v * (g[f] * rsqrtf(1.0f + 1e-5f)) + be[f];
    v = v > 0.0f ? v : 0.0f;
  }
  out[gid] = v;
}

// --------------------------------------------------------------- pooling
__global__ void pool_kernel(const float* __restrict__ emb,
                            const int* __restrict__ batch,
                            float* __restrict__ sums,
                            float* __restrict__ cnts) {
  int gid = blockIdx.x * blockDim.x + threadIdx.x;
  if (gid >= N_NODES * HID) return;
  int node = gid >> 5;
  int f    = gid & 31;
  int bg   = batch[node];
  atomicAdd(&sums[bg * HID + f], emb[gid]);
  if (f == 0) atomicAdd(&cnts[bg], 1.0f);
}

__global__ void pred_kernel(const float* __restrict__ sums,
                            const float* __restrict__ cnts,
                            const float* __restrict__ Wout,
                            const float* __restrict__ bout,
                            float* __restrict__ pred) {
  int gph = blockIdx.x * blockDim.x + threadIdx.x;
  if (gph >= N_GRAPHS) return;
  float cnt = cnts[gph];
  cnt = cnt > 1.0f ? cnt : 1.0f;
  float acc = bout[0];
#pragma unroll
  for (int f = 0; f < HID; ++f)
    acc += (sums[gph * HID + f] / cnt) * Wout[f];
  pred[gph] = 1.0f / (1.0f + __expf(-acc));
}

// ---------------------------------------------------------------- launch
extern "C" void kernel_launch(void* const* d_in, const int* in_sizes, int n_in,
                              void* d_out, int out_size, void* d_ws, size_t ws_size,
                              hipStream_t stream) {
  const float* x     = (const float*)d_in[0];
  const int*   ei    = (const int*)  d_in[1];
  const float* pos   = (const float*)d_in[2];
  const int*   batch = (const int*)  d_in[3];
  const float* s1    = (const float*)d_in[4];
  const float* s2    = (const float*)d_in[5];
  const float* W1 = (const float*)d_in[6];   const float* b1 = (const float*)d_in[7];
  const float* W2 = (const float*)d_in[8];   const float* b2 = (const float*)d_in[9];
  const float* W3 = (const float*)d_in[10];  const float* b3 = (const float*)d_in[11];
  const float* W4 = (const float*)d_in[12];  const float* b4 = (const float*)d_in[13];
  const float* g1 = (const float*)d_in[14];  const float* be1 = (const float*)d_in[15];
  const float* g2 = (const float*)d_in[16];  const float* be2 = (const float*)d_in[17];
  const float* g3 = (const float*)d_in[18];  const float* be3 = (const float*)d_in[19];
  const float* Wout = (const float*)d_in[20];
  const float* bout = (const float*)d_in[21];

  const int* row = ei;
  const int* col = ei + N_EDGES;

  // workspace layout (floats): ~52.5 MB total, L2-resident on MI455X
  float* ws   = (float*)d_ws;
  float* ew   = ws;                               // N_EDGES (becomes coef in place)
  float* deg  = ew   + (size_t)N_EDGES;           // N_NODES
  float* dinv = deg  + (size_t)N_NODES;           // N_NODES
  float* xw   = dinv + (size_t)N_NODES;           // N_NODES*HID
  float* hA   = xw   + (size_t)N_NODES * HID;     // N_NODES*HID
  float* hB   = hA   + (size_t)N_NODES * HID;     // N_NODES*HID
  float* sums = hB   + (size_t)N_NODES * HID;     // N_GRAPHS*HID
  float* cnts = sums + (size_t)N_GRAPHS * HID;    // N_GRAPHS

  float* emb  = (float*)d_out;
  float* pred = (float*)d_out + (size_t)N_NODES * HID;

  const int BLK        = 256;
  const int gNode      = (N_NODES + BLK - 1) / BLK;                 // 391
  const int gEdge      = (N_EDGES + BLK - 1) / BLK;                 // 12500
  const int gNodeFeat  = (N_NODES * HID) / BLK;                     // 12500
  const int gEdgeFeat  = (int)(((long long)N_EDGES * HID) / BLK);   // 400000
  const int gWmma      = (N_NODES / 16 + 3) / 4;                    // 1563 (128-thr blocks)

  // degree / weights / normalization
  fill_zero_kernel<<<gNode, BLK, 0, stream>>>(deg, N_NODES);
  edge_weight_kernel<<<gEdge, BLK, 0, stream>>>(pos, row, col, s1, s2, ew, deg);
  dinv_kernel<<<gNode, BLK, 0, stream>>>(deg, dinv);
  coef_kernel<<<gEdge, BLK, 0, stream>>>(row, col, dinv, ew);

  // ---- layer 1 (K=6 scalar GEMM) -> hA
  gemm_in_kernel<<<gNodeFeat, BLK, 0, stream>>>(x, W1, xw);
  fill_zero_kernel<<<gNodeFeat, BLK, 0, stream>>>(hA, N_NODES * HID);
  scatter_kernel<<<gEdgeFeat, BLK, 0, stream>>>(row, col, ew, xw, hA);
  finalize_kernel<<<gNodeFeat, BLK, 0, stream>>>(hA, xw, dinv, b1, g1, be1, 1, hA);

  // ---- layer 2 (WMMA GEMM) -> hB
  gemm_hid_wmma<<<gWmma, 128, 0, stream>>>(hA, W2, xw);
  fill_zero_kernel<<<gNodeFeat, BLK, 0, stream>>>(hB, N_NODES * HID);
  scatter_kernel<<<gEdgeFeat, BLK, 0, stream>>>(row, col, ew, xw, hB);
  finalize_kernel<<<gNodeFeat, BLK, 0, stream>>>(hB, xw, dinv, b2, g2, be2, 1, hB);

  // ---- layer 3 (WMMA GEMM) -> hA
  gemm_hid_wmma<<<gWmma, 128, 0, stream>>>(hB, W3, xw);
  fill_zero_kernel<<<gNodeFeat, BLK, 0, stream>>>(hA, N_NODES * HID);
  scatter_kernel<<<gEdgeFeat, BLK, 0, stream>>>(row, col, ew, xw, hA);
  finalize_kernel<<<gNodeFeat, BLK, 0, stream>>>(hA, xw, dinv, b3, g3, be3, 1, hA);

  // ---- layer 4 (WMMA GEMM) -> emb (d_out), no BN/ReLU
  gemm_hid_wmma<<<gWmma, 128, 0, stream>>>(hA, W4, xw);
  fill_zero_kernel<<<gNodeFeat, BLK, 0, stream>>>(emb, N_NODES * HID);
  scatter_kernel<<<gEdgeFeat, BLK, 0, stream>>>(row, col, ew, xw, emb);
  finalize_kernel<<<gNodeFeat, BLK, 0, stream>>>(emb, xw, dinv, b4, (const float*)nullptr,
                                                 (const float*)nullptr, 0, emb);

  // ---- pooling + head
  fill_zero_kernel<<<(N_GRAPHS * HID + BLK - 1) / BLK, BLK, 0, stream>>>(sums, N_GRAPHS * HID);
  fill_zero_kernel<<<1, BLK, 0, stream>>>(cnts, N_GRAPHS);
  pool_kernel<<<gNodeFeat, BLK, 0, stream>>>(emb, batch, sums, cnts);
  pred_kernel<<<1, BLK, 0, stream>>>(sums, cnts, Wout, bout, pred);
}